// DQNNetwork_69329362092537
// MI455X (gfx1250) — compile-verified
//
#include <hip/hip_runtime.h>
#include <hip/hip_bf16.h>

#define NSTOCK 65536
#define NFEAT  768
#define NH     512          // H_NUMBER
#define NACT   3

typedef __attribute__((ext_vector_type(2))) float v2f;
typedef __attribute__((ext_vector_type(4))) float v4f;
typedef __attribute__((ext_vector_type(8))) float v8f;

// ---------------------------------------------------------------------------
// Kernel 1: h = relu(x @ Wh^T + bh)   [65536 x 512], K = 768
// WMMA f32 16x16x4, block tile 128x64, BK=32, LDS-staged, 8 waves per block.
// ---------------------------------------------------------------------------
#define BM  128
#define BN  64
#define BK  32
#define LDT 36   // BK + 4 pad: keeps 16B alignment of rows and kills bank conflicts

__global__ __launch_bounds__(256) void k_gemm1(const float* __restrict__ x,
                                               const float* __restrict__ Wh,
                                               const float* __restrict__ bh,
                                               float* __restrict__ h) {
  __shared__ __align__(16) float lx[BM * LDT];
  __shared__ __align__(16) float lw[BN * LDT];

  const int tid  = threadIdx.x;
  const int lane = tid & 31;
  const int wave = tid >> 5;
  const int wm   = (wave & 3) * 32;   // wave row offset inside block tile
  const int wn   = (wave >> 2) * 32;  // wave col offset inside block tile
  const int m0   = blockIdx.x * BM;
  const int n0   = blockIdx.y * BN;

  v8f acc[2][2];
  #pragma unroll
  for (int i = 0; i < 2; ++i)
    #pragma unroll
    for (int j = 0; j < 2; ++j)
      #pragma unroll
      for (int r = 0; r < 8; ++r)
        acc[i][j][r] = 0.0f;

  const int row16 = lane & 15;          // M (for A) / N (for B) index in fragment
  const int koff  = (lane >> 4) * 2;    // lanes 16..31 hold K+2, K+3

  for (int kt = 0; kt < NFEAT; kt += BK) {
    // ---- stage x tile [BM x BK], coalesced float4 (8 threads per row) ----
    #pragma unroll
    for (int it = 0; it < 4; ++it) {
      int idx = tid + it * 256;              // 0..1023
      int r   = idx >> 3;
      int c4  = (idx & 7) << 2;
      v4f v = *(const v4f*)(x + (size_t)(m0 + r) * NFEAT + kt + c4);
      *(v4f*)(lx + r * LDT + c4) = v;
    }
    // ---- stage Wh tile [BN x BK] ----
    #pragma unroll
    for (int it = 0; it < 2; ++it) {
      int idx = tid + it * 256;              // 0..511
      int r   = idx >> 3;
      int c4  = (idx & 7) << 2;
      v4f v = *(const v4f*)(Wh + (size_t)(n0 + r) * NFEAT + kt + c4);
      *(v4f*)(lw + r * LDT + c4) = v;
    }
    __syncthreads();

    // ---- 8 x (4 WMMA) inner K steps ----
    #pragma unroll
    for (int kk = 0; kk < BK; kk += 4) {
      v2f a0 = *(const v2f*)(lx + (wm + row16)      * LDT + kk + koff);
      v2f a1 = *(const v2f*)(lx + (wm + 16 + row16) * LDT + kk + koff);
      v2f b0 = *(const v2f*)(lw + (wn + row16)      * LDT + kk + koff);
      v2f b1 = *(const v2f*)(lw + (wn + 16 + row16) * LDT + kk + koff);
      acc[0][0] = __builtin_amdgcn_wmma_f32_16x16x4_f32(false, a0, false, b0,
                                                        (short)0, acc[0][0], false, false);
      acc[0][1] = __builtin_amdgcn_wmma_f32_16x16x4_f32(false, a0, false, b1,
                                                        (short)0, acc[0][1], false, false);
      acc[1][0] = __builtin_amdgcn_wmma_f32_16x16x4_f32(false, a1, false, b0,
                                                        (short)0, acc[1][0], false, false);
      acc[1][1] = __builtin_amdgcn_wmma_f32_16x16x4_f32(false, a1, false, b1,
                                                        (short)0, acc[1][1], false, false);
    }
    __syncthreads();
  }

  // ---- epilogue: bias + relu, store h ----
  // C/D layout: VGPR r -> M = r (lanes 0-15) or 8+r (lanes 16-31); N = lane&15
  #pragma unroll
  for (int i = 0; i < 2; ++i) {
    const int rbase = m0 + wm + i * 16 + (lane >> 4) * 8;
    #pragma unroll
    for (int j = 0; j < 2; ++j) {
      const int n    = n0 + wn + j * 16 + row16;
      const float bv = bh[n];
      #pragma unroll
      for (int r = 0; r < 8; ++r) {
        float v = acc[i][j][r] + bv;
        v = v > 0.0f ? v : 0.0f;
        h[(size_t)(rbase + r) * NH + n] = v;
      }
    }
  }
}

// ---------------------------------------------------------------------------
// Kernel 2: deterministic column partial sums of h.
// grid (64 row-chunks, 2 col-halves), 256 threads; fully coalesced.
// ---------------------------------------------------------------------------
#define CHUNK_ROWS 1024
#define NCHUNK     (NSTOCK / CHUNK_ROWS)   // 64

__global__ __launch_bounds__(256) void k_colsum(const float* __restrict__ h,
                                                float* __restrict__ Spart) {
  const int c = blockIdx.y * 256 + threadIdx.x;        // 0..511
  const size_t r0 = (size_t)blockIdx.x * CHUNK_ROWS;
  float s = 0.0f;
  for (int r = 0; r < CHUNK_ROWS; ++r)
    s += h[(r0 + r) * NH + c];
  Spart[blockIdx.x * NH + c] = s;
}

// ---------------------------------------------------------------------------
// Kernel 3: fold partials -> S; build We[a][c] = Wf1 - Wf2/(n-1) and
// cst[a] = S . Wf2[a]/(n-1) + bf[a].  Single block of 512 threads.
// ---------------------------------------------------------------------------
__global__ __launch_bounds__(512) void k_prep(const float* __restrict__ Spart,
                                              const float* __restrict__ Wf,
                                              const float* __restrict__ bf,
                                              float* __restrict__ We,
                                              float* __restrict__ cst) {
  __shared__ float red[512];
  const int c = threadIdx.x;               // 0..511
  float S = 0.0f;
  for (int i = 0; i < NCHUNK; ++i)
    S += Spart[i * NH + c];

  const float inv = 1.0f / (float)(NSTOCK - 1);
  float dot[NACT];
  #pragma unroll
  for (int a = 0; a < NACT; ++a) {
    const float w2 = Wf[a * (2 * NH) + NH + c] * inv;
    We[a * NH + c] = Wf[a * (2 * NH) + c] - w2;
    dot[a] = S * w2;
  }
  #pragma unroll
  for (int a = 0; a < NACT; ++a) {
    red[c] = dot[a];
    __syncthreads();
    for (int off = 256; off > 0; off >>= 1) {
      if (c < off) red[c] += red[c + off];
      __syncthreads();
    }
    if (c == 0) cst[a] = red[0] + bf[a];
    __syncthreads();
  }
}

// ---------------------------------------------------------------------------
// Kernel 4: out[n,a] = relu(h[n] . We[a] + cst[a]).
// One wave (32 lanes) per row; lanes split 512 cols into contiguous float4
// chunks (512B per load instruction, fully coalesced); shfl_xor reduction.
// ---------------------------------------------------------------------------
__global__ __launch_bounds__(256) void k_out(const float* __restrict__ h,
                                             const float* __restrict__ We,
                                             const float* __restrict__ cst,
                                             float* __restrict__ out) {
  __shared__ __align__(16) float lwe[NACT * NH];
  __shared__ float lcst[NACT];
  const int tid = threadIdx.x;
  for (int i = tid; i < NACT * NH; i += 256) lwe[i] = We[i];
  if (tid < NACT) lcst[tid] = cst[tid];
  __syncthreads();

  const int lane = tid & 31;
  const int wave = tid >> 5;
  const size_t rowBase = (size_t)blockIdx.x * 64 + (size_t)wave * 8;

  for (int rr = 0; rr < 8; ++rr) {
    const size_t row = rowBase + rr;
    float a0 = 0.0f, a1 = 0.0f, a2 = 0.0f;
    #pragma unroll
    for (int i = 0; i < 4; ++i) {
      const int c = i * 128 + lane * 4;
      v4f hv = *(const v4f*)(h + row * NH + c);
      v4f w0 = *(const v4f*)(lwe + 0 * NH + c);
      v4f w1 = *(const v4f*)(lwe + 1 * NH + c);
      v4f w2 = *(const v4f*)(lwe + 2 * NH + c);
      a0 += hv.x * w0.x + hv.y * w0.y + hv.z * w0.z + hv.w * w0.w;
      a1 += hv.x * w1.x + hv.y * w1.y + hv.z * w1.z + hv.w * w1.w;
      a2 += hv.x * w2.x + hv.y * w2.y + hv.z * w2.z + hv.w * w2.w;
    }
    #pragma unroll
    for (int off = 16; off > 0; off >>= 1) {
      a0 += __shfl_xor(a0, off, 32);
      a1 += __shfl_xor(a1, off, 32);
      a2 += __shfl_xor(a2, off, 32);
    }
    if (lane == 0) {
      out[row * NACT + 0] = fmaxf(a0 + lcst[0], 0.0f);
      out[row * NACT + 1] = fmaxf(a1 + lcst[1], 0.0f);
      out[row * NACT + 2] = fmaxf(a2 + lcst[2], 0.0f);
    }
  }
}

// ---------------------------------------------------------------------------
extern "C" void kernel_launch(void* const* d_in, const int* in_sizes, int n_in,
                              void* d_out, int out_size, void* d_ws, size_t ws_size,
                              hipStream_t stream) {
  const float* x  = (const float*)d_in[0];
  const float* Wh = (const float*)d_in[1];
  const float* bh = (const float*)d_in[2];
  const float* Wf = (const float*)d_in[3];
  const float* bf = (const float*)d_in[4];
  float* out = (float*)d_out;

  char* ws = (char*)d_ws;
  float* h     = (float*)ws;                                  // 65536*512 f32 = 128 MB
  float* Spart = (float*)(ws + (size_t)NSTOCK * NH * 4);      // 64*512
  float* We    = Spart + NCHUNK * NH;                         // 3*512
  float* cst   = We + NACT * NH;                              // 3

  k_gemm1 <<<dim3(NSTOCK / BM, NH / BN), 256, 0, stream>>>(x, Wh, bh, h);
  k_colsum<<<dim3(NCHUNK, 2),           256, 0, stream>>>(h, Spart);
  k_prep  <<<1,                         512, 0, stream>>>(Spart, Wf, bf, We, cst);
  k_out   <<<NSTOCK / 64,               256, 0, stream>>>(h, We, cst, out);
}